// CombinedNetwork_19980187861635
// MI455X (gfx1250) — compile-verified
//
#include <hip/hip_runtime.h>

typedef float v2f __attribute__((ext_vector_type(2)));
typedef float v8f __attribute__((ext_vector_type(8)));

__device__ __forceinline__ v8f splat8(float s) {
    return (v8f){s, s, s, s, s, s, s, s};
}

// ---------------------------------------------------------------------------
// Setup (1 block, 64 threads):
//  (a) collapse each activation-free 2->4->1 MLP into an affine map:
//        p[b,col] = sum_k x[b,k] * Mp[col*8+k] + cp[col],  col = i*8+j
//  (b) build the w row-sum map (affine in x):
//        wsum[b,i] = sum_k x[b,k] * S[i,k] + sb[i]
//        S[i,k] = sum_j Wn[i,j,k],  sb[i] = sum_j bn[i,j]
//      stored as a 16-column WMMA B-tile with column permutation
//        i_col(m) = 2*(m&3) + (m>>3)
//      so lane (lane&7)==t of each j-group holds wsum for i = 2t + (m>>3).
// ---------------------------------------------------------------------------
__global__ void cn_setup(const float* __restrict__ W1, const float* __restrict__ b1,
                         const float* __restrict__ W2, const float* __restrict__ b2,
                         const float* __restrict__ Wn, const float* __restrict__ bn,
                         float* __restrict__ Mp, float* __restrict__ cp,
                         float* __restrict__ Ms, float* __restrict__ sb) {
    int ij = threadIdx.x;
    if (ij >= 64) return;
    {
        int i = ij >> 3, j = ij & 7;
        const float* w1  = W1 + ij * 8;   // [4][2]
        const float* bb1 = b1 + ij * 4;   // [4]
        const float* w2  = W2 + ij * 4;   // [4]
        float a = 0.f, b = 0.f, c = b2[ij];
#pragma unroll
        for (int h = 0; h < 4; ++h) {
            a += w2[h] * w1[h * 2 + 0];
            b += w2[h] * w1[h * 2 + 1];
            c += w2[h] * bb1[h];
        }
        float colv[8];
#pragma unroll
        for (int k = 0; k < 8; ++k) colv[k] = 0.f;
        colv[i] += a;
        colv[j] += b;
#pragma unroll
        for (int k = 0; k < 8; ++k) Mp[ij * 8 + k] = colv[k];
        cp[ij] = c;
    }
    if (ij < 16) {
        const int m = ij;
        const int icol = 2 * (m & 3) + (m >> 3);
#pragma unroll
        for (int k = 0; k < 8; ++k) {
            float s = 0.f;
#pragma unroll
            for (int j = 0; j < 8; ++j) s += Wn[(icol * 8 + j) * 8 + k];
            Ms[m * 8 + k] = s;
        }
        float s = 0.f;
#pragma unroll
        for (int j = 0; j < 8; ++j) s += bn[icol * 8 + j];
        sb[m] = s;
    }
}

// ---------------------------------------------------------------------------
// Main fused kernel: one wave32 per 16 batch rows.
//   p-tiles, w-tiles, and the wsum-tile are all X[16,8] x B[8,16] GEMMs via
//   v_wmma_f32_16x16x4_f32 (K=8 -> 2 WMMAs each, 18 WMMA/wave), biases
//   pre-splatted into C. Only sum_j(p*w) needs a cross-lane reduce
//   (3-step shfl_xor over the 8-lane j-group); wsum comes from the extra
//   WMMA tile so total ds traffic is halved vs. reducing both operands.
//   All output stores are nontemporal (285 MB stream > 192 MB L2).
// ---------------------------------------------------------------------------
__global__ void __launch_bounds__(256)
cn_main(const float* __restrict__ x, const float* __restrict__ Wn,
        const float* __restrict__ bn, const float* __restrict__ Mp,
        const float* __restrict__ cp, const float* __restrict__ Ms,
        const float* __restrict__ sb, float* __restrict__ outp,
        float* __restrict__ outw, float* __restrict__ outres, int nB) {
    const int tid  = blockIdx.x * blockDim.x + threadIdx.x;
    const int wave = tid >> 5;
    const int lane = threadIdx.x & 31;
    const int row0 = wave * 16;
    if (row0 >= nB) return;            // wave-uniform: EXEC stays all-1s
    const int g = lane >> 4;           // K-half / M-half selector
    const int m = lane & 15;           // row (A) / column (B,C,D) within tile

    // A fragments: 16x4 f32, lane holds row m, K = 2g+{0,1}; second chunk K+4
    const float* xr = x + (size_t)(row0 + m) * 8 + g * 2;
    v2f a0 = {xr[0], xr[1]};
    v2f a1 = {xr[4], xr[5]};

    v8f dw[4], dp[4];
#pragma unroll
    for (int t = 0; t < 4; ++t) {
        const int col = t * 16 + m;                 // global output column
        const float* wc = Wn + col * 8 + g * 2;     // Wn[(i*8+o)*8 + k], col-contig
        v2f bw0 = {wc[0], wc[1]};
        v2f bw1 = {wc[4], wc[5]};
        const float* mc = Mp + col * 8 + g * 2;
        v2f bp0 = {mc[0], mc[1]};
        v2f bp1 = {mc[4], mc[5]};

        v8f cw  = splat8(bn[col]);                  // bias rides in as C
        v8f cpv = splat8(cp[col]);
        cw  = __builtin_amdgcn_wmma_f32_16x16x4_f32(false, a0, false, bw0,
                                                    (short)0, cw,  false, false);
        cw  = __builtin_amdgcn_wmma_f32_16x16x4_f32(false, a1, false, bw1,
                                                    (short)0, cw,  false, false);
        cpv = __builtin_amdgcn_wmma_f32_16x16x4_f32(false, a0, false, bp0,
                                                    (short)0, cpv, false, false);
        cpv = __builtin_amdgcn_wmma_f32_16x16x4_f32(false, a1, false, bp1,
                                                    (short)0, cpv, false, false);
        dw[t] = cw;
        dp[t] = cpv;
    }

    // wsum tile: column m holds i_col = 2*(m&3) + (m>>3)
    v8f d5;
    {
        const float* sc = Ms + m * 8 + g * 2;
        v2f b50 = {sc[0], sc[1]};
        v2f b51 = {sc[4], sc[5]};
        d5 = splat8(sb[m]);
        d5 = __builtin_amdgcn_wmma_f32_16x16x4_f32(false, a0, false, b50,
                                                   (short)0, d5, false, false);
        d5 = __builtin_amdgcn_wmma_f32_16x16x4_f32(false, a1, false, b51,
                                                   (short)0, d5, false, false);
    }

#pragma unroll
    for (int t = 0; t < 4; ++t) {
        const int col = t * 16 + m;
        // store p and w: D element (row = row0 + v + 8g, N = col)
#pragma unroll
        for (int v = 0; v < 8; ++v) {
            const int brow = row0 + v + 8 * g;
            const size_t idx = (size_t)brow * 64 + col;
            __builtin_nontemporal_store(dp[t][v], outp + idx);
            __builtin_nontemporal_store(dw[t][v], outw + idx);
        }
        // res[b,i] = (sum_j p*w) / wsum; j == lane&7 within this tile.
        // Butterfly gives the sum to all 8 j-lanes; lane j==t owns the write
        // because its d5 column is exactly wsum[row, i = 2t + (m>>3)].
#pragma unroll
        for (int v = 0; v < 8; ++v) {
            float pw = dp[t][v] * dw[t][v];
#pragma unroll
            for (int s = 1; s < 8; s <<= 1) pw += __shfl_xor(pw, s, 32);
            if ((lane & 7) == t) {
                const int i    = 2 * t + (m >> 3);
                const int brow = row0 + v + 8 * g;
                __builtin_nontemporal_store(pw / d5[v],
                                            outres + (size_t)brow * 8 + i);
            }
        }
    }
}

extern "C" void kernel_launch(void* const* d_in, const int* in_sizes, int n_in,
                              void* d_out, int out_size, void* d_ws, size_t ws_size,
                              hipStream_t stream) {
    const float* x  = (const float*)d_in[0];
    const float* W1 = (const float*)d_in[1];
    const float* b1 = (const float*)d_in[2];
    const float* W2 = (const float*)d_in[3];
    const float* b2 = (const float*)d_in[4];
    const float* Wn = (const float*)d_in[5];
    const float* bn = (const float*)d_in[6];
    const int nB = in_sizes[0] / 8;   // 524288

    float* Mp = (float*)d_ws;         // 512 f32
    float* cp = Mp + 512;             // 64 f32
    float* Ms = cp + 64;              // 128 f32
    float* sb = Ms + 128;             // 16 f32

    float* outp   = (float*)d_out;                    // [B,8,8]
    float* outw   = outp + (size_t)nB * 64;           // [B,8,8]
    float* outres = outp + (size_t)nB * 128;          // [B,8]

    cn_setup<<<1, 64, 0, stream>>>(W1, b1, W2, b2, Wn, bn, Mp, cp, Ms, sb);

    const int waves   = (nB + 15) / 16;
    const int threads = 256;                          // 8 waves per block
    const int blocks  = (waves * 32 + threads - 1) / threads;
    cn_main<<<blocks, threads, 0, stream>>>(x, Wn, bn, Mp, cp, Ms, sb,
                                            outp, outw, outres, nB);
}